// PTM_globalpointer_78082505441812
// MI455X (gfx1250) — compile-verified
//
#include <hip/hip_runtime.h>

typedef __attribute__((ext_vector_type(2))) float v2f;
typedef __attribute__((ext_vector_type(8))) float v8f;

#define WMMA_F32(a, b, c) \
  __builtin_amdgcn_wmma_f32_16x16x4_f32(false, (a), false, (b), (short)0, (c), false, false)

#define BATCH 8
#define SEQ   512
#define HID   768
#define INNER 64
#define N1    128   // 2*inner
#define N2    32    // 2*L
#define NLAB  16
#define NEGC  1000000000000.0f

// ---------------------------------------------------------------------------
// Kernel 1: t = x @ W1 + b1, deinterleave even/odd columns into start/end.
// One block = 16 rows of the 4096-row problem, 8 waves = 8 N-tiles of 16 (N=128).
// ---------------------------------------------------------------------------
__global__ __launch_bounds__(256) void k_gemm1(const float* __restrict__ x,
                                               const float* __restrict__ W1,
                                               const float* __restrict__ b1,
                                               float* __restrict__ st,
                                               float* __restrict__ en) {
  const int lane = threadIdx.x & 31;
  const int wave = threadIdx.x >> 5;          // N tile 0..7
  const int m0   = blockIdx.x * 16;           // row tile over 4096
  const int mA   = m0 + (lane & 15);          // A: lane = M
  const int koff = (lane >> 4) * 2;           // upper half-wave holds K+2,K+3
  const int n    = wave * 16 + (lane & 15);   // B: lane = N

  v8f c = {};
  const float* xrow = x + (size_t)mA * HID + koff;
  for (int k = 0; k < HID; k += 4) {
    v2f a;  a.x  = xrow[k];
            a.y  = xrow[k + 1];
    v2f bb; bb.x = W1[(size_t)(k + koff) * N1 + n];
            bb.y = W1[(size_t)(k + koff + 1) * N1 + n];
    c = WMMA_F32(a, bb, c);
  }

  const float bias = b1[n];
  float* dst   = (n & 1) ? en : st;           // even col -> start, odd -> end
  const int d  = n >> 1;
  const int mB = m0 + ((lane >> 4) << 3);     // C/D: VGPR i = M=i (+8 upper)
#pragma unroll
  for (int i = 0; i < 8; ++i)
    dst[(size_t)(mB + i) * INNER + d] = c[i] + bias;
}

// ---------------------------------------------------------------------------
// Kernel 2: typ[b][c][s] = (x @ W2 + b2)[b,s,c] * 0.5  (stored transposed)
// ---------------------------------------------------------------------------
__global__ __launch_bounds__(64) void k_gemm2(const float* __restrict__ x,
                                              const float* __restrict__ W2,
                                              const float* __restrict__ b2,
                                              float* __restrict__ typ) {
  const int lane = threadIdx.x & 31;
  const int wave = threadIdx.x >> 5;          // N tile 0..1
  const int m0   = blockIdx.x * 16;
  const int mA   = m0 + (lane & 15);
  const int koff = (lane >> 4) * 2;
  const int n    = wave * 16 + (lane & 15);

  v8f c = {};
  const float* xrow = x + (size_t)mA * HID + koff;
  for (int k = 0; k < HID; k += 4) {
    v2f a;  a.x  = xrow[k];
            a.y  = xrow[k + 1];
    v2f bb; bb.x = W2[(size_t)(k + koff) * N2 + n];
            bb.y = W2[(size_t)(k + koff + 1) * N2 + n];
    c = WMMA_F32(a, bb, c);
  }

  const float bias = b2[n];
  const int mB = m0 + ((lane >> 4) << 3);
#pragma unroll
  for (int i = 0; i < 8; ++i) {
    const int r = mB + i;
    const int b = r >> 9;                     // r / 512
    const int s = r & 511;
    typ[((size_t)b * N2 + n) * SEQ + s] = (c[i] + bias) * 0.5f;
  }
}

// ---------------------------------------------------------------------------
// Kernel 3: in-place RoPE on start/end. One thread per (row, freq-pair).
// ---------------------------------------------------------------------------
__global__ __launch_bounds__(256) void k_rope(float* __restrict__ st,
                                              float* __restrict__ en) {
  const int idx = blockIdx.x * blockDim.x + threadIdx.x;   // 0 .. 4096*32-1
  const int r = idx >> 5;          // flat row (b*512+s)
  const int i = idx & 31;          // frequency pair index
  const int s = r & 511;
  const float freq = powf(10000.0f, -(float)(2 * i) / 64.0f);
  const float ang  = (float)s * freq;
  float sn, cs;
  __sincosf(ang, &sn, &cs);

  float* p = st + (size_t)r * INNER + 2 * i;
  float e0 = p[0], e1 = p[1];
  p[0] = e0 * cs - e1 * sn;
  p[1] = e1 * cs + e0 * sn;

  p = en + (size_t)r * INNER + 2 * i;
  e0 = p[0]; e1 = p[1];
  p[0] = e0 * cs - e1 * sn;
  p[1] = e1 * cs + e0 * sn;
}

// ---------------------------------------------------------------------------
// Kernel 4: fused span GEMM + entity-score assembly.
// Block = one 64x64 (m,n) tile of one batch. 16 waves, each a 16x16 WMMA tile
// over K=64. Epilogue broadcasts the span tile across 16 label planes with
// typing row/col terms, attention mask, and strict-lower-triangle -NEG.
// ---------------------------------------------------------------------------
__global__ __launch_bounds__(512) void k_entity(const float* __restrict__ st,
                                                const float* __restrict__ en,
                                                const float* __restrict__ typ,
                                                const int*   __restrict__ mask,
                                                float* __restrict__ out) {
  __shared__ float trowS[NLAB][64];
  __shared__ float tcolS[NLAB][64];
  __shared__ float mrowS[64];
  __shared__ float mcolS[64];

  const int tid = threadIdx.x;
  const int b   = blockIdx.z;
  const int m0  = blockIdx.y * 64;
  const int n0  = blockIdx.x * 64;

  // stage typing rows/cols and mask slices for this tile
  for (int idx = tid; idx < NLAB * 64; idx += 512) {
    const int l = idx >> 6, q = idx & 63;
    trowS[l][q] = typ[((size_t)b * N2 + 2 * l) * SEQ + n0 + q];      // typing[:,0::2] -> +n
    tcolS[l][q] = typ[((size_t)b * N2 + 2 * l + 1) * SEQ + m0 + q];  // typing[:,1::2] -> +m
  }
  if (tid < 64)
    mrowS[tid] = (float)mask[(size_t)b * SEQ + m0 + tid];
  else if (tid < 128)
    mcolS[tid - 64] = (float)mask[(size_t)b * SEQ + n0 + (tid - 64)];
  __syncthreads();

  const int lane = tid & 31;
  const int wave = tid >> 5;                 // 0..15
  const int wm   = wave >> 2;                // m sub-tile 0..3
  const int wn   = wave & 3;                 // n sub-tile 0..3
  const int koff = (lane >> 4) * 2;

  const int mRowA = b * SEQ + m0 + wm * 16 + (lane & 15);   // A lane = M
  const int nRowB = b * SEQ + n0 + wn * 16 + (lane & 15);   // B lane = N
  const float* ap = st + (size_t)mRowA * INNER + koff;
  const float* bp = en + (size_t)nRowB * INNER + koff;

  v8f c = {};
#pragma unroll
  for (int k = 0; k < INNER; k += 4) {
    v2f a;  a.x  = ap[k];  a.y  = ap[k + 1];
    v2f bb; bb.x = bp[k];  bb.y = bp[k + 1];
    c = WMMA_F32(a, bb, c);
  }

  const int nn  = wn * 16 + (lane & 15);     // local n within 64
  const int n_g = n0 + nn;
  const float mc = mcolS[nn];

#pragma unroll
  for (int l = 0; l < NLAB; ++l) {
    const float tr = trowS[l][nn];
    float* plane = out + ((size_t)(b * NLAB + l) * SEQ) * SEQ;
#pragma unroll
    for (int i = 0; i < 8; ++i) {
      const int mm  = wm * 16 + i + ((lane >> 4) << 3);
      const int m_g = m0 + mm;
      float v = c[i] * 0.125f + tr + tcolS[l][mm];          // span/sqrt(64) + typing terms
      const float mr = mrowS[mm];
      v = v * mr - NEGC * (1.0f - mr);
      v = v * mc - NEGC * (1.0f - mc);
      if (m_g > n_g) v -= NEGC;                             // tril(k=-1)
      plane[(size_t)m_g * SEQ + n_g] = v;
    }
  }
}

// ---------------------------------------------------------------------------
extern "C" void kernel_launch(void* const* d_in, const int* in_sizes, int n_in,
                              void* d_out, int out_size, void* d_ws, size_t ws_size,
                              hipStream_t stream) {
  const float* x    = (const float*)d_in[0];
  const float* W1   = (const float*)d_in[1];
  const float* b1   = (const float*)d_in[2];
  const float* W2   = (const float*)d_in[3];
  const float* b2   = (const float*)d_in[4];
  const int*   am   = (const int*)d_in[5];
  float* out = (float*)d_out;

  float* ws  = (float*)d_ws;
  float* st  = ws;                                  // 4096*64
  float* en  = ws + (size_t)BATCH * SEQ * INNER;    // 4096*64
  float* typ = en + (size_t)BATCH * SEQ * INNER;    // 8*32*512

  k_gemm1<<<dim3(BATCH * SEQ / 16), dim3(256), 0, stream>>>(x, W1, b1, st, en);
  k_gemm2<<<dim3(BATCH * SEQ / 16), dim3(64), 0, stream>>>(x, W2, b2, typ);
  k_rope<<<dim3(BATCH * SEQ * (INNER / 2) / 256), dim3(256), 0, stream>>>(st, en);
  k_entity<<<dim3(SEQ / 64, SEQ / 64, BATCH), dim3(512), 0, stream>>>(st, en, typ, am, out);
}